// DiscriminationLoss_85487029059989
// MI455X (gfx1250) — compile-verified
//
#include <hip/hip_runtime.h>
#include <hip/hip_bf16.h>
#include <math.h>

// ---------------------------------------------------------------------------
// DiscriminationLoss on MI455X (gfx1250)
//   Stage 1: bandwidth-bound segment reduction (144 MB @ 23.3 TB/s ~= 6.2 us)
//            f32 LDS atomics, per-wave privatized, bank-spread stride-9 layout
//   Stage 2: 32x32 pairwise A[i]+A[j] built with V_WMMA_F32_16X16X4_F32,
//            then log(max(sigma - sqrt, 0)^2 + 1) upper-triangle reduction.
// ---------------------------------------------------------------------------

#define NCH 8
#define NLAB 33                    // labels 0..32 (0 = background)
#define ACC_STRIDE 9               // 8 channel sums + 1 count; 9 coprime to 64 banks
#define ACC_WORDS (NLAB * ACC_STRIDE)  // 297 floats
#define WAVES_PER_BLOCK 8
#define BLOCK 256
#define GRID 512
#define SIGMA_DIS 3.0f

typedef __attribute__((ext_vector_type(2))) float v2f;
typedef __attribute__((ext_vector_type(8))) float v8f;

// --- Stage 0: zero the 297-word accumulator in workspace (every call) ------
__global__ __launch_bounds__(512) void zero_ws_kernel(float* __restrict__ g) {
  int i = threadIdx.x;
  if (i < ACC_WORDS) g[i] = 0.0f;
}

// --- Stage 1: segment sums S[k,c] and counts N[k] --------------------------
__global__ __launch_bounds__(BLOCK) void seg_reduce_kernel(
    const float* __restrict__ pred,     // [8][P] channel-major
    const int* __restrict__ labels,     // [P]
    float* __restrict__ gacc,           // [297]
    int P) {
  __shared__ float acc[WAVES_PER_BLOCK][ACC_WORDS];
  const int tid = threadIdx.x;
  const int wave = tid >> 5;  // wave32

  for (int i = tid; i < WAVES_PER_BLOCK * ACC_WORDS; i += BLOCK)
    (&acc[0][0])[i] = 0.0f;
  __syncthreads();

  float* __restrict__ my = acc[wave];
  const int stride = GRID * BLOCK * 4;
  for (int p = (blockIdx.x * BLOCK + tid) * 4; p < P; p += stride) {
    const int4 lab = *(const int4*)(labels + p);
    __builtin_prefetch(labels + p + stride, 0, 0);  // global_prefetch_b8
    float4 v[NCH];
#pragma unroll
    for (int c = 0; c < NCH; ++c)
      v[c] = *(const float4*)(pred + (size_t)c * (size_t)P + p);

    const int b0 = lab.x * ACC_STRIDE;
    const int b1 = lab.y * ACC_STRIDE;
    const int b2 = lab.z * ACC_STRIDE;
    const int b3 = lab.w * ACC_STRIDE;
#pragma unroll
    for (int c = 0; c < NCH; ++c) {
      atomicAdd(&my[b0 + c], v[c].x);   // ds_add_f32
      atomicAdd(&my[b1 + c], v[c].y);
      atomicAdd(&my[b2 + c], v[c].z);
      atomicAdd(&my[b3 + c], v[c].w);
    }
    atomicAdd(&my[b0 + NCH], 1.0f);
    atomicAdd(&my[b1 + NCH], 1.0f);
    atomicAdd(&my[b2 + NCH], 1.0f);
    atomicAdd(&my[b3 + NCH], 1.0f);
  }
  __syncthreads();

  for (int i = tid; i < ACC_WORDS; i += BLOCK) {
    float s = 0.0f;
#pragma unroll
    for (int w = 0; w < WAVES_PER_BLOCK; ++w) s += acc[w][i];
    atomicAdd(&gacc[i], s);             // global_atomic_add_f32
  }
}

// --- Stage 2: pairwise loss via V_WMMA_F32_16X16X4_F32 ---------------------
// pair[i][j] = A[i] + A[j] built as two chained rank-broadcast WMMAs:
//   D  = (A_i/4) * ones   (every K slot of A-operand holds A_row/4, B all 1)
//   D += (1/4)  * A_j     (A-operand all 1/4, every K slot of B holds A_col)
// Replicating across all K slots makes this independent of the exact K->lane
// packing; only row = lane%16 (A) and col = lane%16 (B/C/D) are relied upon.
__global__ __launch_bounds__(32) void pair_loss_kernel(
    const float* __restrict__ gacc, float* __restrict__ out) {
  __shared__ float As[32];
  const int lane = threadIdx.x;  // one full wave32, EXEC all ones

  // A[k] = N[k+1] * sum_c S[k+1][c]^2   (background label 0 dropped)
  const float* row = gacc + (lane + 1) * ACC_STRIDE;
  float q = 0.0f;
#pragma unroll
  for (int c = 0; c < NCH; ++c) { float s = row[c]; q += s * s; }
  As[lane] = row[NCH] * q;
  __syncthreads();

  const int m = lane & 15;
  const int half = lane >> 4;
  float total = 0.0f;
#pragma unroll
  for (int ti = 0; ti < 2; ++ti) {
#pragma unroll
    for (int tj = 0; tj < 2; ++tj) {
      v2f a1; a1[0] = a1[1] = As[ti * 16 + m] * 0.25f;  // row i broadcast
      v2f b1; b1[0] = b1[1] = 1.0f;
      v8f c = {};
      c = __builtin_amdgcn_wmma_f32_16x16x4_f32(
          false, a1, false, b1, (short)0, c, false, false);
      v2f a2; a2[0] = a2[1] = 0.25f;
      v2f b2; b2[0] = b2[1] = As[tj * 16 + m];          // col j broadcast
      c = __builtin_amdgcn_wmma_f32_16x16x4_f32(
          false, a2, false, b2, (short)0, c, false, false);
      // C/D layout: VGPR r, lanes 0-15 -> M=r, lanes 16-31 -> M=r+8; N=lane%16
#pragma unroll
      for (int r = 0; r < 8; ++r) {
        const int i = ti * 16 + r + 8 * half;
        const int j = tj * 16 + m;
        float d = sqrtf(c[r]);
        float dd = fmaxf(SIGMA_DIS - d, 0.0f);
        float term = logf(dd * dd + 1.0f);
        total += (i < j) ? term : 0.0f;  // select, keeps EXEC uniform
      }
    }
  }
#pragma unroll
  for (int off = 16; off > 0; off >>= 1)
    total += __shfl_down(total, off, 32);
  if (lane == 0) out[0] = total * (31.0f / 32.0f);  // (K-1)/K
}

// ---------------------------------------------------------------------------
extern "C" void kernel_launch(void* const* d_in, const int* in_sizes, int n_in,
                              void* d_out, int out_size, void* d_ws, size_t ws_size,
                              hipStream_t stream) {
  const float* pred  = (const float*)d_in[0];  // [8, 2048, 2048] f32
  // d_in[1] = regions_mask (unused by reference forward)
  const int* labels  = (const int*)d_in[2];    // [2048, 2048] i32
  // d_in[3] = num_kernel scalar (K=32, compile-time constant here)
  float* gacc = (float*)d_ws;                  // 297 floats of scratch
  float* out  = (float*)d_out;                 // 1 float
  const int P = in_sizes[2];                   // H*W = 4,194,304

  zero_ws_kernel<<<1, 512, 0, stream>>>(gacc);
  seg_reduce_kernel<<<GRID, BLOCK, 0, stream>>>(pred, labels, gacc, P);
  pair_loss_kernel<<<1, 32, 0, stream>>>(gacc, out);
}